// NeuralODE_69166153334984
// MI455X (gfx1250) — compile-verified
//
#include <hip/hip_runtime.h>

typedef __attribute__((ext_vector_type(2))) float v2f;
typedef __attribute__((ext_vector_type(8))) float v8f;

// ---------------- Tsit5 tableau (f32, matches reference) ----------------
__device__ const float TS_A[7][6] = {
  { 0.f, 0.f, 0.f, 0.f, 0.f, 0.f },
  { 0.161f, 0.f, 0.f, 0.f, 0.f, 0.f },
  { -0.008480655492356989f, 0.335480655492357f, 0.f, 0.f, 0.f, 0.f },
  { 2.8971530571054935f, -6.359448489975075f, 4.3622954328695815f, 0.f, 0.f, 0.f },
  { 5.325864828439257f, -11.748883564062828f, 7.4955393428898365f, -0.09249506636175525f, 0.f, 0.f },
  { 5.86145544294642f, -12.92096931784711f, 8.159367898576159f, -0.071584973281401f, -0.028269050394068383f, 0.f },
  // stage 6 input is y_new (B coefficients); its eval is the FSAL k7
  { 0.09646076681806523f, 0.01f, 0.4798896504144996f, 1.379008574103742f, -3.290069515436081f, 2.324710524099774f },
};

#define E1f (-0.00178001105222577714f)
#define E2f (-0.0008164344596567469f)
#define E3f ( 0.007880878010261995f)
#define E4f (-0.1447110071732629f)
#define E5f ( 0.5823571654525552f)
#define E6f (-0.45808210592918697f)
#define E7f ( 0.015151515151515152f)

#define RTOL_  1e-3f
#define ATOL_  1e-6f
#define SAFE_  0.9f
#define FMIN_  0.2f
#define FMAX_  10.0f

// ---------------- dynamic-LDS workspace (one 16-trajectory batch) --------
// B-fragment-packed weights: WBb[l][kp][n] = { WT[2kp][n], WT[2kp+1][n] },
// i.e. the two K values one lane needs per fragment are adjacent (1x b64).
// Activations likewise: A[m][kp] = { a[m][2kp], a[m][2kp+1] }.
struct SMem {
  float  W0b[32][2]; float b0b[32]; float dW0[32][2]; float db0[32]; // layer0 raw
  float2 WBb[2][16][32]; float2 WBd[2][16][32];                     // hidden l=0,1
  float  bb[2][32];      float  db[2][32];
  float2 WBh[16][16];    float2 WBdh[16][16];                       // head (N padded to 16)
  float  bbh[16];        float  dbh[16];
  float2 A[16][16];      float2 dA[16][16];                         // 16 samples x 32 feats
  float  Y[16][2];
  float  K[7][16][2];
};

// Re-declared in every function so InferAddressSpaces sees the addrspace(3)
// root locally -> plain ds_load/ds_store with immediate offsets, no generic
// pointer null-checks across the __noinline__ boundaries.
__device__ __forceinline__ SMem& smem() {
  extern __shared__ char smraw[];
  return *reinterpret_cast<SMem*>(smraw);
}

// ---------------- f32 WMMA 16x16x4 wrapper -------------------------------
__device__ __forceinline__ v8f wmma4(v2f a, v2f b, v8f c) {
  return __builtin_amdgcn_wmma_f32_16x16x4_f32(
      /*neg_a=*/false, a, /*neg_b=*/false, b,
      /*c_mod=*/(short)0, c, /*reuse_a=*/false, /*reuse_b=*/false);
}

// softplus (jax: logaddexp(z,0)) + sigmoid-scaled tangent, numerically stable
__device__ __forceinline__ void act_pair(float z, float dz, float* a, float* da) {
  float e  = expf(-fabsf(z));
  *a  = fmaxf(z, 0.f) + log1pf(e);
  float sg = ((z >= 0.f) ? 1.f : e) / (1.f + e);
  *da = sg * dz;
}

// ---------------- hidden layer: (A,dA) <- layer_l(A,dA), 48 WMMAs --------
__device__ __noinline__ void hidden_layer(int l) {
  SMem& sm = smem();
  const int lane = threadIdx.x & 31;
  const int half = lane >> 4;       // 0: lanes 0-15, 1: lanes 16-31
  const int lm   = lane & 15;
  v8f z0 = {}, z1 = {}, dz0 = {}, dz1 = {};
#pragma unroll
  for (int c = 0; c < 8; ++c) {                 // K = 32 in chunks of 4
    const int kp = 2 * c + half;                // fragment k-pair index
    const float2 aAf = sm.A[lm][kp];            // A-frag: M=lm, K=2kp..2kp+1
    const float2 aDf = sm.dA[lm][kp];
    const float2 w0 = sm.WBb[l][kp][lm];        // B-frag: N=lm (+16)
    const float2 w1 = sm.WBb[l][kp][16 + lm];
    const float2 d0 = sm.WBd[l][kp][lm];
    const float2 d1 = sm.WBd[l][kp][16 + lm];
    v2f aA  = { aAf.x, aAf.y }, aD  = { aDf.x, aDf.y };
    v2f bW0 = { w0.x, w0.y },   bW1 = { w1.x, w1.y };
    v2f bD0 = { d0.x, d0.y },   bD1 = { d1.x, d1.y };
    z0  = wmma4(aA, bW0, z0);   z1  = wmma4(aA, bW1, z1);   // z  = a @ Wb^T
    dz0 = wmma4(aD, bW0, dz0);  dz1 = wmma4(aD, bW1, dz1);  // dz += da @ Wb^T
    dz0 = wmma4(aA, bD0, dz0);  dz1 = wmma4(aA, bD1, dz1);  // dz += a @ dW^T
  }
  __syncthreads();   // all reads of A/dA complete before overwrite
  const float bbl0 = sm.bb[l][lm], bbl1 = sm.bb[l][16 + lm];
  const float dbl0 = sm.db[l][lm], dbl1 = sm.db[l][16 + lm];
  float* Af  = reinterpret_cast<float*>(&sm.A[0][0]);
  float* dAf = reinterpret_cast<float*>(&sm.dA[0][0]);
#pragma unroll
  for (int r = 0; r < 8; ++r) {                 // C-layout: M=r+8*half, N=lm
    const int m = r + 8 * half;
    float a, da;
    act_pair(z0[r] + bbl0, dz0[r] + dbl0, &a, &da);
    Af[m * 32 + lm] = a;        dAf[m * 32 + lm] = da;
    act_pair(z1[r] + bbl1, dz1[r] + dbl1, &a, &da);
    Af[m * 32 + 16 + lm] = a;   dAf[m * 32 + 16 + lm] = da;
  }
  __syncthreads();
}

// ---------------- full vector-field eval: K[s] = f(Y), 16 samples --------
__device__ __noinline__ void mlp_eval(int s) {
  SMem& sm = smem();
  const int lane = threadIdx.x & 31;
  const int half = lane >> 4;
  const int lm   = lane & 15;

  // layer 0 (K=2) in VALU, writing A/dA directly in [m][feature] layout
  float* Af  = reinterpret_cast<float*>(&sm.A[0][0]);
  float* dAf = reinterpret_cast<float*>(&sm.dA[0][0]);
#pragma unroll
  for (int t = 0; t < 2; ++t) {
    const int n = t * 16 + lm;
    const float w0 = sm.W0b[n][0], w1 = sm.W0b[n][1], b  = sm.b0b[n];
    const float u0 = sm.dW0[n][0], u1 = sm.dW0[n][1], ub = sm.db0[n];
#pragma unroll
    for (int r = 0; r < 8; ++r) {
      const int m = r + 8 * half;
      const float x0 = sm.Y[m][0], x1 = sm.Y[m][1];
      float a, da;
      act_pair(w0 * x0 + w1 * x1 + b, u0 * x0 + u1 * x1 + ub, &a, &da);
      Af[m * 32 + n] = a;  dAf[m * 32 + n] = da;
    }
  }
  __syncthreads();

  hidden_layer(0);
  hidden_layer(1);

  // head: one 16x16 tile (N 0..1 valid, weights zero-padded), 24 WMMAs
  v8f z = {}, dz = {};
#pragma unroll
  for (int c = 0; c < 8; ++c) {
    const int kp = 2 * c + half;
    const float2 aAf = sm.A[lm][kp],  aDf = sm.dA[lm][kp];
    const float2 wf  = sm.WBh[kp][lm], df = sm.WBdh[kp][lm];
    v2f aA = { aAf.x, aAf.y }, aD = { aDf.x, aDf.y };
    v2f bW = { wf.x, wf.y },   bD = { df.x, df.y };
    z  = wmma4(aA, bW, z);
    dz = wmma4(aD, bW, dz);
    dz = wmma4(aA, bD, dz);
  }
  if (lm < 2) {
    const float bsum = sm.bbh[lm] + sm.dbh[lm];
#pragma unroll
    for (int r = 0; r < 8; ++r)
      sm.K[s][r + 8 * half][lm] = z[r] + dz[r] + bsum;   // out = base + JVP
  }
}

// ---------------- kernel: one block = one (field, 16-trajectory batch) ---
struct KParams {
  const float* x0s; const float* tev;
  const float* w1[4]; const float* b1[4];
  const float* w2[4]; const float* b2[4];
  float* out;
};

__global__ __launch_bounds__(32)
void node_tsit5_kernel(KParams p) {
  SMem& sm = smem();
  const int tid   = threadIdx.x;
  const int field = blockIdx.x >> 5;   // 0: f1 (base=p2, tangent=p1-p2), 1: f2
  const int batch = blockIdx.x & 31;

  const float *BW[4], *BB[4], *OW[4], *OB[4];
#pragma unroll
  for (int l = 0; l < 4; ++l) {
    if (field == 0) { BW[l] = p.w2[l]; BB[l] = p.b2[l]; OW[l] = p.w1[l]; OB[l] = p.b1[l]; }
    else            { BW[l] = p.w1[l]; BB[l] = p.b1[l]; OW[l] = p.w2[l]; OB[l] = p.b2[l]; }
  }

  // ---- stage weights + tangents (other - base) into LDS ----
  for (int i = tid; i < 64; i += 32) {            // layer 0 W (32x2)
    const int n = i >> 1, k = i & 1;
    const float wb = BW[0][i];
    sm.W0b[n][k] = wb;  sm.dW0[n][k] = OW[0][i] - wb;
  }
  {                                               // 32-wide biases
    const int i = tid;
    float b;
    b = BB[0][i]; sm.b0b[i]  = b; sm.db0[i]   = OB[0][i] - b;
    b = BB[1][i]; sm.bb[0][i] = b; sm.db[0][i] = OB[1][i] - b;
    b = BB[2][i]; sm.bb[1][i] = b; sm.db[1][i] = OB[2][i] - b;
  }
  {                                               // hidden W -> packed frags
    float* WBbF = reinterpret_cast<float*>(&sm.WBb[0][0][0]);
    float* WBdF = reinterpret_cast<float*>(&sm.WBd[0][0][0]);
    for (int i = tid; i < 1024; i += 32) {        // W[n][k], row-major n*32+k
      const int n = i >> 5, k = i & 31;
      const int kp = k >> 1, ke = k & 1;
      const int o1 = (((0 * 16 + kp) * 32) + n) * 2 + ke;
      const int o2 = (((1 * 16 + kp) * 32) + n) * 2 + ke;
      float w;
      w = BW[1][i]; WBbF[o1] = w; WBdF[o1] = OW[1][i] - w;
      w = BW[2][i]; WBbF[o2] = w; WBdF[o2] = OW[2][i] - w;
    }
  }
  {                                               // head W (2x32) -> [16][16] f2 padded
    float* WBhF  = reinterpret_cast<float*>(&sm.WBh[0][0]);
    float* WBdhF = reinterpret_cast<float*>(&sm.WBdh[0][0]);
    for (int i = tid; i < 512; i += 32) {
      const int k = i >> 4, n = i & 15;
      const int kp = k >> 1, ke = k & 1;
      const float wb = (n < 2) ? BW[3][n * 32 + k] : 0.f;
      const float wo = (n < 2) ? OW[3][n * 32 + k] : 0.f;
      const int o = ((kp * 16) + n) * 2 + ke;
      WBhF[o] = wb;  WBdhF[o] = wo - wb;
    }
  }
  if (tid < 16) {
    const float b = (tid < 2) ? BB[3][tid] : 0.f;
    const float o = (tid < 2) ? OB[3][tid] : 0.f;
    sm.bbh[tid] = b;  sm.dbh[tid] = o - b;
  }
  __syncthreads();

  // ---- per-trajectory state on lanes 0..15 ----
  const int sample = batch * 16 + (tid & 15);
  const size_t obase = (size_t)field * 10240 + (size_t)sample * 20;
  float y0r = 0.f, y1r = 0.f, dtr = 1e-3f;
  if (tid < 16) {
    y0r = p.x0s[2 * sample];  y1r = p.x0s[2 * sample + 1];
    p.out[obase + 0] = y0r;   p.out[obase + 1] = y1r;    // save index 0 = y0
  }

#pragma unroll 1
  for (int iv = 0; iv < 9; ++iv) {
    float tv = 0.f, t1v = 0.f;  bool done = false;
    if (tid < 16) { tv = p.tev[iv];  t1v = p.tev[iv + 1]; }

#pragma unroll 1
    for (int st = 0; st < 40; ++st) {
      float h = 0.f;
      if (tid < 16) h = fminf(dtr, t1v - tv);
      float yn0 = 0.f, yn1 = 0.f;

#pragma unroll 1
      for (int s = 0; s < 7; ++s) {               // 7 stages, one eval copy
        if (tid < 16) {
          float ax = y0r, ay = y1r;
          for (int j = 0; j < s; ++j) {
            const float cf = TS_A[s][j];
            ax += h * cf * sm.K[j][tid][0];
            ay += h * cf * sm.K[j][tid][1];
          }
          if (s == 6) { yn0 = ax; yn1 = ay; }     // stage 6 input == y_new
          sm.Y[tid][0] = ax;  sm.Y[tid][1] = ay;
        }
        __syncthreads();
        mlp_eval(s);
        __syncthreads();
      }

      if (tid < 16) {
        float e0 = E1f * sm.K[0][tid][0] + E2f * sm.K[1][tid][0]
                 + E3f * sm.K[2][tid][0] + E4f * sm.K[3][tid][0]
                 + E5f * sm.K[4][tid][0] + E6f * sm.K[5][tid][0]
                 + E7f * sm.K[6][tid][0];
        float e1 = E1f * sm.K[0][tid][1] + E2f * sm.K[1][tid][1]
                 + E3f * sm.K[2][tid][1] + E4f * sm.K[3][tid][1]
                 + E5f * sm.K[4][tid][1] + E6f * sm.K[5][tid][1]
                 + E7f * sm.K[6][tid][1];
        e0 *= h;  e1 *= h;
        const float sc0 = ATOL_ + RTOL_ * fmaxf(fabsf(y0r), fabsf(yn0));
        const float sc1 = ATOL_ + RTOL_ * fmaxf(fabsf(y1r), fabsf(yn1));
        const float r0 = e0 / sc0, r1 = e1 / sc1;
        const float errn = sqrtf(0.5f * (r0 * r0 + r1 * r1));
        const bool accept = (errn <= 1.0f);
        float factor = (errn > 0.f) ? SAFE_ * powf(errn, -0.2f) : FMAX_;
        factor = fminf(fmaxf(factor, FMIN_), FMAX_);
        if (accept && !done) { tv += h;  y0r = yn0;  y1r = yn1; }
        if (!done) dtr = fmaxf(h * factor, 1e-10f);
        done = done || ((t1v - tv) <= 1e-9f);
      }
    }

    if (tid < 16) {
      p.out[obase + (size_t)(iv + 1) * 2 + 0] = y0r;
      p.out[obase + (size_t)(iv + 1) * 2 + 1] = y1r;
    }
  }
}

extern "C" void kernel_launch(void* const* d_in, const int* in_sizes, int n_in,
                              void* d_out, int out_size, void* d_ws, size_t ws_size,
                              hipStream_t stream) {
  (void)in_sizes; (void)n_in; (void)d_ws; (void)ws_size; (void)out_size;
  KParams p;
  p.x0s = (const float*)d_in[0];
  p.tev = (const float*)d_in[1];
  // setup_inputs dict order: per layer l -> w1_l, b1_l, w2_l, b2_l
  for (int l = 0; l < 4; ++l) {
    p.w1[l] = (const float*)d_in[2 + 4 * l];
    p.b1[l] = (const float*)d_in[3 + 4 * l];
    p.w2[l] = (const float*)d_in[4 + 4 * l];
    p.b2[l] = (const float*)d_in[5 + 4 * l];
  }
  p.out = (float*)d_out;
  // 2 fields x 32 batches of 16 trajectories; one wave32 per block,
  // whole solve runs out of 27 KB of LDS per WGP-resident workgroup.
  node_tsit5_kernel<<<dim3(64), dim3(32), sizeof(SMem), stream>>>(p);
}